// dec_contrast_53334903881829
// MI455X (gfx1250) — compile-verified
//
#include <hip/hip_runtime.h>
#include <hip/hip_bf16.h>
#include <math.h>

typedef __attribute__((ext_vector_type(2))) float v2f;
typedef __attribute__((ext_vector_type(8))) float v8f;

// ---- problem constants (from reference) ----
#define KCLS   19           // NUM_CLASSES
#define KPAD   32           // padded class count for 2x 16-row WMMA tiles
#define CCH    256          // INNER (feature channels)
#define QLEN   2975         // queue length
#define HW     16384        // 128*128
#define NPOS   131072       // 8*128*128 positions
#define RES_ELEMS 2490368   // 8*19*128*128
#define INV_TEMP 5.0f       // 1/0.2

#define CHUNK  32           // positions per LDS tile
#define NCHUNKS (NPOS / CHUNK)
#define CHPAD  36           // LDS row stride (floats): 16B-aligned rows,
                            // banks 4*((ch*9)%16)+j -> conflict-free B reads
#define SEG_GRID 256        // workgroups for the segsum kernel

// ---- workspace layout (float offsets) ----
#define WS_SUMS   0                    // KPAD*CCH = 8192
#define WS_COUNTS (WS_SUMS + KPAD*CCH) // 32
#define WS_QN     (WS_COUNTS + KPAD)   // 8192 (only 19*256 used)
#define WS_LOSS   (WS_QN + KPAD*CCH)   // 1
#define WS_TOTAL  (WS_LOSS + 1)        // 16417 floats (~66 KB)

__global__ void dc_init_kernel(float* __restrict__ ws) {
    int i = blockIdx.x * blockDim.x + threadIdx.x;
    if (i < WS_TOTAL) ws[i] = 0.0f;
}

// Scatter-sum of fea by argmax class, expressed as onehot x fea GEMM using
// full-precision V_WMMA_F32_16X16X4_F32 (exact: A entries are 0/1, f32 acc).
__global__ __launch_bounds__(512)
void dc_segsum_wmma_kernel(const float* __restrict__ fea,
                           const float* __restrict__ res,
                           float* __restrict__ ws) {
    __shared__ float sfea[CCH * CHPAD];   // [channel][pos] tile, padded stride
    __shared__ int   spred[CHUNK];
    __shared__ int   shist[KPAD];

    const int tid  = threadIdx.x;
    const int lane = tid & 31;
    const int w    = tid >> 5;       // wave id 0..15 -> channel tile
    const int col  = lane & 15;      // N column within tile / A row within tile
    const int half = lane >> 4;      // 0: K=0,1 ; 1: K=2,3 (A and B layouts)
    const int ch   = w * 16 + col;   // this lane's channel column

    if (tid < KPAD) shist[tid] = 0;

    v8f acc0 = {};  // classes 0..15
    v8f acc1 = {};  // classes 16..31 (only 16..18 can be nonzero)

    for (int ci = blockIdx.x; ci < NCHUNKS; ci += SEG_GRID) {
        const int n0  = ci * CHUNK;
        const int b   = n0 >> 14;          // batch (CHUNK divides HW)
        const int hw0 = n0 & (HW - 1);
        __syncthreads();
        // per-position argmax over 19 classes + local histogram
        if (tid < CHUNK) {
            const float* rp = res + (((size_t)b * KCLS) << 14) + hw0 + tid;
            float bv = rp[0]; int bi = 0;
            #pragma unroll
            for (int k = 1; k < KCLS; ++k) {
                float v = rp[(size_t)k << 14];
                if (v > bv) { bv = v; bi = k; }
            }
            spred[tid] = bi;
            atomicAdd(&shist[bi], 1);
        }
        // coalesced 128-bit fea tile staging: [256 channels][32 positions]
        const float* fb = fea + (((size_t)b * CCH) << 14) + hw0;
        #pragma unroll
        for (int idx = tid; idx < CCH * CHUNK / 4; idx += 512) {
            const int c  = idx >> 3;            // 8 float4 per channel row
            const int j4 = (idx & 7) << 2;
            const float4 v = *(const float4*)(fb + (((size_t)c) << 14) + j4);
            *(float4*)(&sfea[c * CHPAD + j4]) = v;
        }
        // prefetch next grid-stride chunk: one 128B cacheline (= one channel
        // row of 32 floats) per thread -> global_prefetch_b8
        {
            const int ci2 = ci + SEG_GRID;
            if (ci2 < NCHUNKS && tid < CCH) {
                const int n2  = ci2 * CHUNK;
                const int b2  = n2 >> 14;
                const int hw2 = n2 & (HW - 1);
                __builtin_prefetch(fea + (((size_t)b2 * CCH + tid) << 14) + hw2, 0, 3);
            }
        }
        __syncthreads();
        #pragma unroll
        for (int step = 0; step < CHUNK / 4; ++step) {
            const int j0 = step * 4 + (half ? 2 : 0);
            const int j1 = j0 + 1;
            v2f bm;
            bm.x = sfea[ch * CHPAD + j0];
            bm.y = sfea[ch * CHPAD + j1];
            const int p0 = spred[j0];
            const int p1 = spred[j1];
            v2f a0, a1;
            a0.x = (p0 == col)      ? 1.0f : 0.0f;
            a0.y = (p1 == col)      ? 1.0f : 0.0f;
            a1.x = (p0 == col + 16) ? 1.0f : 0.0f;
            a1.y = (p1 == col + 16) ? 1.0f : 0.0f;
            acc0 = __builtin_amdgcn_wmma_f32_16x16x4_f32(
                false, a0, false, bm, (short)0, acc0, false, false);
            acc1 = __builtin_amdgcn_wmma_f32_16x16x4_f32(
                false, a1, false, bm, (short)0, acc1, false, false);
        }
    }
    // flush accumulators: C/D layout -> VGPR i holds M = i (+8 for lanes>=16)
    float* sums = ws + WS_SUMS;
    #pragma unroll
    for (int i = 0; i < 8; ++i) {
        const int m = i + (half ? 8 : 0);
        atomicAdd(&sums[(size_t)m * CCH + ch],        acc0[i]);
        atomicAdd(&sums[(size_t)(m + 16) * CCH + ch], acc1[i]);
    }
    __syncthreads();
    if (tid < KPAD) atomicAdd(&ws[WS_COUNTS + tid], (float)shist[tid]);
}

// keys = sums / max(count,1); qn = keys / max(||keys||, 1e-12)
__global__ __launch_bounds__(256)
void dc_keys_kernel(float* __restrict__ ws) {
    __shared__ float red[256];
    const int k = blockIdx.x;
    const int c = threadIdx.x;
    const float cnt = ws[WS_COUNTS + k];
    const float key = ws[WS_SUMS + (size_t)k * CCH + c] / fmaxf(cnt, 1.0f);
    red[c] = key * key;
    for (int off = 128; off > 0; off >>= 1) {
        __syncthreads();
        if (c < off) red[c] += red[c + off];
    }
    __syncthreads();
    const float nrm = sqrtf(red[0]);
    ws[WS_QN + (size_t)k * CCH + c] = key / fmaxf(nrm, 1e-12f);
}

// One block per (k,c) row: build qsum row in LDS (queues stay L2-resident),
// then online logsumexp over 2*QLEN logits; accumulate loss via f32 atomic.
__global__ __launch_bounds__(256)
void dc_lse_kernel(const float* __restrict__ queues, float* __restrict__ ws) {
    __shared__ float sq[QLEN];
    __shared__ float redm[256];
    __shared__ float reds[256];
    const int k   = blockIdx.x >> 8;     // / 256
    const int c   = blockIdx.x & 255;
    const int tid = threadIdx.x;
    const float a = ws[WS_QN + (size_t)k * CCH + c];
    const float* xr = queues + (size_t)k * (CCH * (size_t)QLEN) + (size_t)c * QLEN;
    // qsum[c, :] = sum_k queues[k, c, :]
    for (int q = tid; q < QLEN; q += 256) {
        const float* qp = queues + (size_t)c * QLEN + q;
        float s = 0.0f;
        #pragma unroll
        for (int kk = 0; kk < KCLS; ++kk)
            s += qp[(size_t)kk * (CCH * (size_t)QLEN)];
        sq[q] = s;
    }
    __syncthreads();
    float m = -INFINITY, ssum = 0.0f;
    for (int q = tid; q < QLEN; q += 256) {
        const float x   = xr[q];
        const float pos = a * x * INV_TEMP;
        const float neg = a * (sq[q] - x) * INV_TEMP;
        float nm = fmaxf(m, pos);
        ssum = ssum * __expf(m - nm) + __expf(pos - nm);
        m = nm;
        nm = fmaxf(m, neg);
        ssum = ssum * __expf(m - nm) + __expf(neg - nm);
        m = nm;
    }
    redm[tid] = m; reds[tid] = ssum;
    for (int off = 128; off > 0; off >>= 1) {
        __syncthreads();
        if (tid < off) {
            const float m1 = redm[tid], m2 = redm[tid + off];
            const float s1 = reds[tid], s2 = reds[tid + off];
            const float nm = fmaxf(m1, m2);
            reds[tid] = s1 * __expf(m1 - nm) + s2 * __expf(m2 - nm);
            redm[tid] = nm;
        }
    }
    __syncthreads();
    if (tid == 0 && ws[WS_COUNTS + k] > 0.0f) {
        const float lse  = redm[0] + __logf(reds[0]);
        const float pos0 = a * xr[0] * INV_TEMP;
        atomicAdd(&ws[WS_LOSS], (lse - pos0) * (1.0f / 256.0f));
    }
}

__global__ void dc_write_loss_kernel(const float* __restrict__ ws,
                                     float* __restrict__ out) {
    out[RES_ELEMS] = ws[WS_LOSS];
}

extern "C" void kernel_launch(void* const* d_in, const int* in_sizes, int n_in,
                              void* d_out, int out_size, void* d_ws, size_t ws_size,
                              hipStream_t stream) {
    const float* fea    = (const float*)d_in[0];
    const float* res    = (const float*)d_in[1];
    const float* queues = (const float*)d_in[2];
    float* out = (float*)d_out;
    float* ws  = (float*)d_ws;
    (void)in_sizes; (void)n_in; (void)out_size; (void)ws_size;

    // output[0:RES_ELEMS] = res (first tuple element), graph-capture safe D2D
    hipMemcpyAsync(out, res, sizeof(float) * RES_ELEMS,
                   hipMemcpyDeviceToDevice, stream);

    dc_init_kernel<<<(WS_TOTAL + 255) / 256, 256, 0, stream>>>(ws);
    dc_segsum_wmma_kernel<<<SEG_GRID, 512, 0, stream>>>(fea, res, ws);
    dc_keys_kernel<<<KCLS, 256, 0, stream>>>(ws);
    dc_lse_kernel<<<KCLS * CCH, 256, 0, stream>>>(queues, ws);
    dc_write_loss_kernel<<<1, 1, 0, stream>>>(ws, out);
}